// TorchMatrixLogm_81389630259672
// MI455X (gfx1250) — compile-verified
//
#include <hip/hip_runtime.h>
#include <math.h>

// ---------------------------------------------------------------------------
// Matrix logarithm, inverse scaling-and-squaring, all-GEMM formulation.
// Core op: f32 WMMA GEMM (v_wmma_f32_16x16x4_f32).
// LDS staging priority: TDM tensor_load_to_lds (TENSORcnt) >
//                       per-lane async global->LDS (ASYNCcnt) > plain loads.
// ---------------------------------------------------------------------------

typedef float v2f  __attribute__((ext_vector_type(2)));
typedef float v4f  __attribute__((ext_vector_type(4)));
typedef float v8f  __attribute__((ext_vector_type(8)));
typedef int   v4i  __attribute__((ext_vector_type(4)));
typedef unsigned int u32x4 __attribute__((ext_vector_type(4)));
typedef int   i32x8 __attribute__((ext_vector_type(8)));
typedef int   i32x4 __attribute__((ext_vector_type(4)));

#define TILE_M 128
#define TILE_N 128
#define TILE_K 32
#define APAD   4     // pad A rows in LDS: stride 36 dwords breaks bank aliasing

#if defined(__HIP_DEVICE_COMPILE__) && __has_builtin(__builtin_amdgcn_tensor_load_to_lds) && __has_builtin(__builtin_amdgcn_s_wait_tensorcnt)
#define USE_TDM 1
#else
#define USE_TDM 0
#endif

#if defined(__HIP_DEVICE_COMPILE__) && __has_builtin(__builtin_amdgcn_global_load_async_to_lds_b128)
#define USE_ASYNC_LDS 1
#else
#define USE_ASYNC_LDS 0
#endif

#if defined(__HIP_DEVICE_COMPILE__)
typedef __attribute__((address_space(1))) v4i* as1_v4i;   // global int4*
typedef __attribute__((address_space(3))) v4i* as3_v4i;   // LDS int4*
typedef __attribute__((address_space(3))) void* as3_void;
#if __has_builtin(__builtin_amdgcn_s_wait_asynccnt)
#define WAIT_ASYNC(n) __builtin_amdgcn_s_wait_asynccnt(n)
#else
#define WAIT_ASYNC(n) asm volatile("s_wait_asynccnt %0" ::"i"(n) : "memory")
#endif
#else
#define WAIT_ASYNC(n) ((void)0)
#endif

#if USE_TDM
// Issue one TDM 2D tile load: tile_h rows x tile_w f32 elements from a row-major
// n-stride matrix at gsrc into LDS at lds_byte_addr. pad_ctrl adds LDS row
// padding (load only). D# bit layout per CDNA5 ISA ch.8.
__device__ __forceinline__ void tdm_load_tile(const float* gsrc,
                                              unsigned lds_byte_addr, int n,
                                              unsigned tile_w, unsigned tile_h,
                                              unsigned pad_ctrl) {
  unsigned long long ga = (unsigned long long)(size_t)gsrc;
  u32x4 g0;
  g0.x = 1u;                                            // count=1, user desc
  g0.y = lds_byte_addr;                                 // lds_addr [63:32]
  g0.z = (unsigned)ga;                                  // global_addr lo
  g0.w = (unsigned)((ga >> 32) & 0x1FFFFFFu) | (2u << 30);  // addr hi | type=2
  unsigned un = (unsigned)n;
  i32x8 g1;
  g1[0] = (int)((2u << 16) | pad_ctrl);                 // data_size=4B | pad cfg
  g1[1] = (int)((un & 0xFFFFu) << 16);                  // tensor_dim0[15:0]
  g1[2] = (int)((un >> 16) | ((un & 0xFFFFu) << 16));   // dim0 hi | dim1 lo
  g1[3] = (int)((un >> 16) | (tile_w << 16));           // dim1 hi | tile_dim0
  g1[4] = (int)tile_h;                                  // tile_dim1 | tile_dim2=0
  g1[5] = (int)un;                                      // dim0_stride[31:0]
  g1[6] = 0;                                            // stride hi | dim1_stride lo
  g1[7] = 0;
  i32x4 z4 = {0, 0, 0, 0};
  i32x8 z8 = {0, 0, 0, 0, 0, 0, 0, 0};
  // amdgpu-toolchain (clang-23) 6-arg form:
  // (u32x4 g0, i32x8 g1, i32x4, i32x4, i32x8, i32 cpol)
  __builtin_amdgcn_tensor_load_to_lds(g0, g1, z4, z4, z8, 0);
}
#endif

// C = alpha * A * B   (row-major, n x n, n multiple of 128)
__global__ __launch_bounds__(256) void logm_gemm_wmma_f32(
    const float* __restrict__ A, const float* __restrict__ B,
    float* __restrict__ C, float alpha, int n) {
  __shared__ float As[2][TILE_M][TILE_K + APAD];  // 2 x 18 KB
  __shared__ float Bs[2][TILE_K][TILE_N];         // 2 x 16 KB

  const int tid   = threadIdx.x;
  const int lane  = tid & 31;       // wave32
  const int wave  = tid >> 5;       // 8 waves
  const int wm    = wave & 1;       // 2 waves along M (64 rows each)
  const int wn    = wave >> 1;      // 4 waves along N (32 cols each)
  const int ln16  = lane & 15;
  const int khalf = lane >> 4;      // 0 -> K{0,1}, 1 -> K{2,3} (f32 frag layout)

  const int bm = blockIdx.y * TILE_M;
  const int bn = blockIdx.x * TILE_N;

  // per-thread staging coordinates (non-TDM paths)
  const int arow0 = tid >> 3;             // 8 float4 per 32-float A row
  const int ac4   = (tid & 7) << 2;
  const int brow0 = tid >> 5;             // 32 float4 per 128-float B row
  const int bc4   = (tid & 31) << 2;

  auto stage = [&](int buf, int k0g) {
#if USE_TDM
    // wave 0 issues two tile DMAs; TDM pad recreates the 36-dword A row stride.
    unsigned lds_a = (unsigned)(size_t)(as3_void)&As[buf][0][0];
    unsigned lds_b = (unsigned)(size_t)(as3_void)&Bs[buf][0][0];
    // pad: every 32 dwords (interval code 4) insert 4 dwords (amount code 3)
    const unsigned padA = (1u << 20) | (4u << 22) | (3u << 25);
    tdm_load_tile(&A[(size_t)bm * n + k0g], lds_a, n, TILE_K, TILE_M, padA);
    tdm_load_tile(&B[(size_t)k0g * n + bn], lds_b, n, TILE_N, TILE_K, 0u);
#elif USE_ASYNC_LDS
    #pragma unroll
    for (int t = 0; t < 4; ++t) {
      int row = arow0 + t * 32;
      const float* gp = &A[(size_t)(bm + row) * n + k0g + ac4];
      __builtin_amdgcn_global_load_async_to_lds_b128(
          (as1_v4i)(v4i*)const_cast<float*>(gp),
          (as3_v4i)(v4i*)&As[buf][row][ac4], 0, 0);
    }
    #pragma unroll
    for (int t = 0; t < 4; ++t) {
      int row = brow0 + t * 8;
      const float* gp = &B[(size_t)(k0g + row) * n + bn + bc4];
      __builtin_amdgcn_global_load_async_to_lds_b128(
          (as1_v4i)(v4i*)const_cast<float*>(gp),
          (as3_v4i)(v4i*)&Bs[buf][row][bc4], 0, 0);
    }
#else
    #pragma unroll
    for (int t = 0; t < 4; ++t) {
      int row = arow0 + t * 32;
      *(v4f*)(&As[buf][row][ac4]) =
          *(const v4f*)(&A[(size_t)(bm + row) * n + k0g + ac4]);
    }
    #pragma unroll
    for (int t = 0; t < 4; ++t) {
      int row = brow0 + t * 8;
      *(v4f*)(&Bs[buf][row][bc4]) =
          *(const v4f*)(&B[(size_t)(k0g + row) * n + bn + bc4]);
    }
    __builtin_prefetch(&A[(size_t)(bm + arow0) * n + k0g + TILE_K + ac4], 0, 0);
    __builtin_prefetch(&B[(size_t)(k0g + TILE_K + brow0) * n + bn + bc4], 0, 0);
#endif
  };

  v8f acc[4][2];
  #pragma unroll
  for (int mt = 0; mt < 4; ++mt)
    #pragma unroll
    for (int nt = 0; nt < 2; ++nt)
      acc[mt][nt] = (v8f){0.f, 0.f, 0.f, 0.f, 0.f, 0.f, 0.f, 0.f};

  const int nk = n / TILE_K;

#if USE_TDM
  if (wave == 0) stage(0, 0);           // uniform branch: only wave 0 issues
#else
  stage(0, 0);
#endif

  for (int kt = 0; kt < nk; ++kt) {
    const int cur = kt & 1;
#if USE_TDM
    // wave 0: queue next tile's 2 DMAs, then drain down to those 2 (TDM ops
    // from one wave complete in order -> current buffer is resident).
    if (wave == 0) {
      if (kt + 1 < nk) {
        stage(cur ^ 1, (kt + 1) * TILE_K);
        __builtin_amdgcn_s_wait_tensorcnt(2);
      } else {
        __builtin_amdgcn_s_wait_tensorcnt(0);
      }
    }
#else
    if (kt + 1 < nk) {
      stage(cur ^ 1, (kt + 1) * TILE_K);
      WAIT_ASYNC(8);
    } else {
      WAIT_ASYNC(0);
    }
#endif
    __syncthreads();

    // ---- compute: 8 K-steps of 4, 8 WMMA each = 64 v_wmma per tile ---------
    #pragma unroll
    for (int k0 = 0; k0 < TILE_K; k0 += 4) {
      v2f af[4];
      #pragma unroll
      for (int mt = 0; mt < 4; ++mt) {
        int row = wm * 64 + mt * 16 + ln16;
        af[mt] = *(const v2f*)(&As[cur][row][k0 + 2 * khalf]);
      }
      v2f bf[2];
      #pragma unroll
      for (int nt = 0; nt < 2; ++nt) {
        int col = wn * 32 + nt * 16 + ln16;
        bf[nt].x = Bs[cur][k0 + 2 * khalf][col];
        bf[nt].y = Bs[cur][k0 + 2 * khalf + 1][col];
      }
      #pragma unroll
      for (int mt = 0; mt < 4; ++mt)
        #pragma unroll
        for (int nt = 0; nt < 2; ++nt)
          acc[mt][nt] = __builtin_amdgcn_wmma_f32_16x16x4_f32(
              false, af[mt], false, bf[nt], (short)0, acc[mt][nt], false, false);
    }
    __syncthreads();  // protect 'cur' buffer before it is restaged next iter
  }

  // ---- epilogue: C/D layout — vgpr r: lanes0-15 M=r, lanes16-31 M=r+8 ------
  #pragma unroll
  for (int mt = 0; mt < 4; ++mt) {
    #pragma unroll
    for (int nt = 0; nt < 2; ++nt) {
      int m0  = bm + wm * 64 + mt * 16 + khalf * 8;
      int col = bn + wn * 32 + nt * 16 + ln16;
      #pragma unroll
      for (int r = 0; r < 8; ++r)
        C[(size_t)(m0 + r) * n + col] = alpha * acc[mt][nt][r];
    }
  }
}

// out = cx*X + cI*I   (elementwise; covers scale / identity / shift)
__global__ void logm_comb(float* __restrict__ out, const float* __restrict__ X,
                          float cx, float cI, int n) {
  int idx = blockIdx.x * blockDim.x + threadIdx.x;
  int total = n * n;
  if (idx >= total) return;
  int row = idx / n;
  int col = idx - row * n;
  float v = cx * X[idx];
  if (row == col) v += cI;
  out[idx] = v;
}

// out += a*X
__global__ void logm_axpy(float* __restrict__ out, const float* __restrict__ X,
                          float a, int total) {
  int idx = blockIdx.x * blockDim.x + threadIdx.x;
  if (idx >= total) return;
  out[idx] += a * X[idx];
}

// ---------------------------------------------------------------------------
extern "C" void kernel_launch(void* const* d_in, const int* in_sizes, int n_in,
                              void* d_out, int out_size, void* d_ws, size_t ws_size,
                              hipStream_t stream) {
  const float* dA = (const float*)d_in[0];
  float* out = (float*)d_out;

  int total = in_sizes[0];
  int n = 1;
  while (n * n < total) n <<= 1;            // n = 2048
  const size_t MAT = (size_t)n * n;

  // 9 scratch matrices in d_ws
  float* base = (float*)d_ws;
  float* CUR = base + 0 * MAT;
  float* Y   = base + 1 * MAT;
  float* Z   = base + 2 * MAT;
  float* P   = base + 3 * MAT;
  float* W   = base + 4 * MAT;
  float* T1  = base + 5 * MAT;
  float* MI  = base + 6 * MAT;
  float* XI  = base + 7 * MAT;
  float* R   = base + 8 * MAT;

  dim3 ggrid(n / TILE_N, n / TILE_M);
  int ew_blocks = (int)((MAT + 255) / 256);

  auto gemm = [&](const float* a, const float* b, float* c, float alpha) {
    logm_gemm_wmma_f32<<<ggrid, 256, 0, stream>>>(a, b, c, alpha, n);
  };
  auto comb = [&](float* o, const float* x, float cx, float cI) {
    logm_comb<<<ew_blocks, 256, 0, stream>>>(o, x, cx, cI, n);
  };
  auto axpy = [&](float* o, const float* x, float a) {
    logm_axpy<<<ew_blocks, 256, 0, stream>>>(o, x, a, (int)MAT);
  };

  const int NUM_SQRTM = 8;
  const int NS_SQRT_ITERS = 12;
  const int NS_INV_ITERS  = 14;

  // Gauss-Legendre order-8 nodes/weights (leggauss(8))
  const double gl_node[8] = {-0.9602898564975363, -0.7966664774136267,
                             -0.5255324099163290, -0.1834346424956498,
                              0.1834346424956498,  0.5255324099163290,
                              0.7966664774136267,  0.9602898564975363};
  const double gl_wt[8]   = { 0.1012285362903763,  0.2223810344533745,
                              0.3137066458778873,  0.3626837833783620,
                              0.3626837833783620,  0.3137066458778873,
                              0.2223810344533745,  0.1012285362903763};

  // ---- B = A^(1/2^8) via 8 coupled Newton-Schulz square roots --------------
  const float* curIn = dA;
  for (int s = 0; s < NUM_SQRTM; ++s) {
    // spectral prescale: A eigs in [1,5] -> /3 puts them in (0,2) for NS;
    // later roots have spectra near 1.
    float c = (s == 0) ? 3.0f : 1.5f;
    comb(Y, curIn, 1.0f / c, 0.0f);   // Y0 = M/c
    comb(Z, curIn, 0.0f, 1.0f);       // Z0 = I
    float *y = Y, *z = Z, *p = P, *t = T1;
    for (int it = 0; it < NS_SQRT_ITERS; ++it) {
      gemm(z, y, p, 1.0f);            // P = Z*Y
      comb(W, p, -0.5f, 1.5f);        // W = 0.5*(3I - P)
      gemm(y, W, t, 1.0f);            // Ynew = Y*W
      gemm(W, z, p, 1.0f);            // Znew = W*Z   (overwrites consumed P)
      float* tmp;
      tmp = y; y = t; t = tmp;
      tmp = z; z = p; p = tmp;
    }
    comb(CUR, y, sqrtf(c), 0.0f);     // sqrt(M) = sqrt(c) * Ylim
    curIn = CUR;
  }

  // ---- Pade quadrature: out = sum_i alpha_i * inv((1-b)I + b*B) * (B - I) --
  comb(R, curIn, 1.0f, -1.0f);        // R = B - I
  comb(out, R, 0.0f, 0.0f);           // out = 0

  for (int i = 0; i < 8; ++i) {
    float beta  = (float)((gl_node[i] + 1.0) * 0.5);
    float alphi = (float)(gl_wt[i] * 128.0);          // w * 2^(s-1)
    comb(MI, curIn, beta, 1.0f - beta);               // M_i, eigs ~ 1
    comb(XI, MI, 0.0f, 1.0f);                         // X0 = I
    float *x = XI, *t = T1;
    for (int it = 0; it < NS_INV_ITERS; ++it) {
      gemm(MI, x, P, 1.0f);           // P = M*X
      comb(W, P, -1.0f, 2.0f);        // W = 2I - P
      gemm(x, W, t, 1.0f);            // Xnew = X*W
      float* tmp = x; x = t; t = tmp;
    }
    gemm(x, R, P, 1.0f);              // P = M^-1 * R
    axpy(out, P, alphi);              // out += alpha_i * P
  }
}